// IF_53120155517152
// MI455X (gfx1250) — compile-verified
//
#include <hip/hip_runtime.h>

// IF (integrate-and-fire) forward scan, T=4, B=8, FEAT=1024x3072, f32.
// Pure HBM-streaming elementwise recurrence -> bandwidth-bound (~1 GB traffic,
// ~40us floor at 23.3 TB/s). WMMA is inapplicable (no contraction). Optimized
// for the CDNA5 memory path: 128-bit global loads/stores, NT temporal hints on
// the single-touch streams (x, out), RT (default) on the L2-reused parameters
// (thresh2, dtmem), all 4 strided time-step loads issued before the dependent
// recurrence, wave32-friendly 256-thread blocks.

typedef __attribute__((ext_vector_type(4))) float v4f;

#define T_STEPS   4
#define BATCH     8
#define FEAT_VEC  (1024 * 3072 / 4)        /* 786432 float4 per feature map   */
#define SLAB_VEC  (BATCH * FEAT_VEC)       /* 6291456 float4 per timestep     */
#define BLOCK     256

__global__ __launch_bounds__(BLOCK) void if_fwd_kernel(
    const float* __restrict__ x,        // [T*B, 1024, 3072]
    const float* __restrict__ thresh2,  // [1024, 3072]
    const float* __restrict__ dtmem,    // [1024, 3072]
    float* __restrict__ out)            // [T*B, 1024, 3072]
{
    // grid: (FEAT_VEC/BLOCK, BATCH). fv = float4 index into the feature map,
    // v = float4 index into one timestep slab. Both fit in 32 bits.
    const unsigned fv = blockIdx.x * BLOCK + threadIdx.x;     // < FEAT_VEC
    const unsigned v  = blockIdx.y * (unsigned)FEAT_VEC + fv; // < SLAB_VEC

    const v4f* xv  = (const v4f*)x;
    const v4f* thv = (const v4f*)thresh2;
    const v4f* dmv = (const v4f*)dtmem;
    v4f*       ov  = (v4f*)out;

    // Parameters: default (RT) temporal hint -> stay hot in the 192MB L2,
    // reused by all 8 batch rows.
    v4f th  = thv[fv];
    v4f dm  = dmv[fv];
    v4f mem = dm * th;                      // mem0 = dtmem * thre

    // Issue all 4 time-step loads (100MB apart) up front: one wait covers the
    // whole dependent chain. NT hint: each byte of x is read exactly once.
    v4f xr[T_STEPS];
#pragma unroll
    for (int t = 0; t < T_STEPS; ++t)
        xr[t] = __builtin_nontemporal_load(xv + (size_t)t * SLAB_VEC + v);

#pragma unroll
    for (int t = 0; t < T_STEPS; ++t) {
        v4f s;
#pragma unroll
        for (int c = 0; c < 4; ++c) {       // clang vector subscript: rw-legal
            float m  = mem[c] + xr[t][c];                 // integrate
            float sp = (m >= th[c]) ? th[c] : 0.0f;       // heaviside * thre
            mem[c]   = m - sp;                            // soft reset
            s[c]     = sp;
        }
        // Spikes are written once and not re-read: NT store, don't pollute L2.
        __builtin_nontemporal_store(s, ov + (size_t)t * SLAB_VEC + v);
    }
}

extern "C" void kernel_launch(void* const* d_in, const int* in_sizes, int n_in,
                              void* d_out, int out_size, void* d_ws, size_t ws_size,
                              hipStream_t stream) {
    (void)in_sizes; (void)n_in; (void)out_size; (void)d_ws; (void)ws_size;

    const float* x       = (const float*)d_in[0];  // [T*B,1024,3072] f32
    const float* thresh2 = (const float*)d_in[1];  // [1024,3072] f32
    const float* dtmem   = (const float*)d_in[2];  // [1024,3072] f32
    float*       out     = (float*)d_out;          // [T*B,1024,3072] f32

    dim3 grid(FEAT_VEC / BLOCK, BATCH, 1);         // 3072 x 8 blocks, exact
    dim3 block(BLOCK, 1, 1);                       // 8 wave32 per block
    if_fwd_kernel<<<grid, block, 0, stream>>>(x, thresh2, dtmem, out);
}